// resample_69312182223188
// MI455X (gfx1250) — compile-verified
//
#include <hip/hip_runtime.h>

// Bilinear resample (deformable grid sample), MI455X / gfx1250.
// Wave-per-pixel, float4-per-lane, 4-deep async gather-to-LDS pipeline.

typedef float v4f __attribute__((ext_vector_type(4)));

#define WVS   8        // waves per block (256 threads)
#define PPW   16       // pixels processed per wave
#define DEPTH 4        // pipeline stages (LDS buffers per wave)

static constexpr int kS  = 128;         // spatial size
static constexpr int kC  = 128;         // channels
static constexpr int kCB = kC * 4;      // bytes per pixel-vector (512)

__global__ __launch_bounds__(WVS * 32)
void bilinear_resample_kernel(const float* __restrict__ offsets,
                              const float* __restrict__ inputs,
                              float* __restrict__ out)
{
    // Staging: per wave, DEPTH buffers x 4 corners x 128 floats = 8 KB -> 64 KB/block.
    __shared__ __align__(16) float sh[WVS * DEPTH * 4 * kC];

    const unsigned lane = threadIdx.x & 31u;
    const unsigned wave = threadIdx.x >> 5;

    // Wave-uniform flat wave index -> scalar (SMEM) loads of offsets and
    // SALU address generation (verified in round-1 asm: s_load_b64 + readfirstlane).
    const unsigned waveFlat =
        (unsigned)__builtin_amdgcn_readfirstlane((int)(blockIdx.x * WVS + wave));
    const unsigned pix0 = waveFlat * PPW;

    // LDS byte offset of this wave's staging area (low 32 bits of a generic
    // pointer to LDS == byte offset within the workgroup's LDS allocation).
    const unsigned ldsWave =
        (unsigned)(size_t)(&sh[wave * DEPTH * 4 * kC]) + lane * 16u;

    float  w00[DEPTH], w01[DEPTH], w10[DEPTH], w11[DEPTH];
    size_t outoff[DEPTH];

    #pragma unroll
    for (int k = 0; k < PPW + (DEPTH - 1); ++k) {
        // ---- issue stage: gather pixel k's 4 corner vectors into slot k%DEPTH
        if (k < PPW) {
            const unsigned p   = pix0 + (unsigned)k;
            const unsigned b   = p >> 14;        // / (128*128)
            const unsigned rem = p & 16383u;
            const unsigned i   = rem >> 7;
            const unsigned j   = rem & 127u;

            // coords = clip(offset + grid, 0, 127)
            const float2 off = *(const float2*)(offsets + 2ull * p);
            const float fi = fminf(fmaxf(off.x + (float)i, 0.0f), 127.0f);
            const float fj = fminf(fmaxf(off.y + (float)j, 0.0f), 127.0f);
            const float fl_i = floorf(fi), fl_j = floorf(fj);
            const int lt_i = (int)fl_i,      lt_j = (int)fl_j;
            const int rb_i = (int)ceilf(fi), rb_j = (int)ceilf(fj);
            const float fx = fi - fl_i;   // fraction along i (rows)
            const float fy = fj - fl_j;   // fraction along j (cols)

            const int slot = k & (DEPTH - 1);
            w00[slot] = (1.0f - fx) * (1.0f - fy);   // (lt_i, lt_j)
            w01[slot] = (1.0f - fx) * fy;            // (lt_i, rb_j)
            w10[slot] = fx * (1.0f - fy);            // (rb_i, lt_j)
            w11[slot] = fx * fy;                     // (rb_i, rb_j)
            outoff[slot] = (size_t)p * kC + lane * 4u;

            const float* base = inputs + (size_t)b * (kS * kS * kC);
            const float* a00 = base + ((size_t)(lt_i * kS + lt_j)) * kC + lane * 4u;
            const float* a01 = base + ((size_t)(lt_i * kS + rb_j)) * kC + lane * 4u;
            const float* a10 = base + ((size_t)(rb_i * kS + lt_j)) * kC + lane * 4u;
            const float* a11 = base + ((size_t)(rb_i * kS + rb_j)) * kC + lane * 4u;

            // Async gather of the 4 corner vectors into LDS (ASYNCcnt += 4).
            const unsigned l = ldsWave + (unsigned)slot * (4 * kCB);
            asm volatile("global_load_async_to_lds_b128 %0, %1, off"
                         :: "v"(l + 0 * kCB), "v"((unsigned long long)a00) : "memory");
            asm volatile("global_load_async_to_lds_b128 %0, %1, off"
                         :: "v"(l + 1 * kCB), "v"((unsigned long long)a01) : "memory");
            asm volatile("global_load_async_to_lds_b128 %0, %1, off"
                         :: "v"(l + 2 * kCB), "v"((unsigned long long)a10) : "memory");
            asm volatile("global_load_async_to_lds_b128 %0, %1, off"
                         :: "v"(l + 3 * kCB), "v"((unsigned long long)a11) : "memory");
        }

        // ---- compute stage: consume pixel kc = k - (DEPTH-1)
        const int kc = k - (DEPTH - 1);
        if (kc >= 0) {
            // Async loads complete in order. Groups newer than kc still in
            // flight: (issued_total - kc - 1); wait until only those remain.
            const int issued = (k + 1 < PPW) ? (k + 1) : PPW;
            const int pending = 4 * (issued - kc - 1);   // constant after unroll
            asm volatile("s_wait_asynccnt %0" :: "i"(pending) : "memory");

            const int slot = kc & (DEPTH - 1);
            const float* s = &sh[(wave * DEPTH + slot) * 4 * kC];
            const v4f v00 = *(const v4f*)(s + 0 * kC + lane * 4u);
            const v4f v01 = *(const v4f*)(s + 1 * kC + lane * 4u);
            const v4f v10 = *(const v4f*)(s + 2 * kC + lane * 4u);
            const v4f v11 = *(const v4f*)(s + 3 * kC + lane * 4u);

            v4f r = v00 * w00[slot] + v01 * w01[slot]
                  + v10 * w10[slot] + v11 * w11[slot];

            // Output is streamed once (134 MB): non-temporal so the 128 MB
            // input (4x reuse) stays resident in the 192 MB L2.
            __builtin_nontemporal_store(r, (v4f*)(out + outoff[slot]));
        }
    }
}

extern "C" void kernel_launch(void* const* d_in, const int* in_sizes, int n_in,
                              void* d_out, int out_size, void* d_ws, size_t ws_size,
                              hipStream_t stream) {
    const float* offsets = (const float*)d_in[0];   // (B, S, S, 2) f32
    const float* inputs  = (const float*)d_in[1];   // (B, S, S, C) f32
    float* out = (float*)d_out;                     // (B, S, S, C) f32

    const int total_pix     = in_sizes[0] / 2;      // B*S*S = 262144
    const int pix_per_block = WVS * PPW;            // 128
    dim3 grid((unsigned)((total_pix + pix_per_block - 1) / pix_per_block));
    dim3 block(WVS * 32);                           // 256 threads = 8 waves

    bilinear_resample_kernel<<<grid, block, 0, stream>>>(offsets, inputs, out);
}